// FullAttention_87247965651107
// MI455X (gfx1250) — compile-verified
//
#include <hip/hip_runtime.h>

// MI455X / gfx1250, wave32. FP32 WMMA path: V_WMMA_F32_16X16X4_F32.
typedef float v2f __attribute__((ext_vector_type(2)));
typedef float v8f __attribute__((ext_vector_type(8)));

#define B_ 2
#define H_ 8
#define N_ 4096
#define D_ 64
#define SCALE 0.125f          // 1/sqrt(64)
#define NEG_BIG (-1.0e30f)    // "-inf" stand-in; expf underflows to exact 0

__launch_bounds__(256, 2)
__global__ void attn_fp32_wmma_kernel(const float* __restrict__ Q,
                                      const float* __restrict__ K,
                                      const float* __restrict__ V,
                                      const int*   __restrict__ mask,
                                      float* __restrict__ Xout,
                                      float* __restrict__ Aout) {
  __shared__ float red[8][32];               // per-wave {M[16], L[16]}

  const int tid  = threadIdx.x;
  const int w    = tid >> 5;                 // wave in WG (wave32)
  const int lane = tid & 31;
  const int hi   = lane >> 4;                // half-wave index (0/1)
  const int lo   = lane & 15;

  const int bh    = blockIdx.x >> 5;         // (b*H + h)
  const int qt    = blockIdx.x & 31;         // q-tile group of 128 rows
  const int b     = bh >> 3;                 // H_ == 8
  const int qbase = qt * 128 + w * 16;       // this wave's 16 query rows

  const size_t bhND = (size_t)bh * N_ * D_;
  const float* Qb = Q + bhND;
  const float* Kb = K + bhND;
  const float* Vb = V + bhND;
  float* Abase = Aout + ((size_t)bh * N_ + qbase) * N_;  // 16 attn rows of this wave

  // ---------- load Q tile as 16 A-fragments (16x4 f32 WMMA layout) ----------
  // lane: row = lo, cols = 4t + 2*hi + {0,1}
  v2f qf[16];
  {
    const float* qp = Qb + (size_t)(qbase + lo) * D_ + 2 * hi;
#pragma unroll
    for (int t = 0; t < 16; ++t)
      qf[t] = *(const v2f*)(qp + 4 * t);
  }

  // ---------- pass 1: S = Q K^T * scale, mask, store raw s, online (M,L) ----------
  float Mv[8], Lv[8];
#pragma unroll
  for (int r = 0; r < 8; ++r) { Mv[r] = NEG_BIG; Lv[r] = 0.0f; }

  for (int j = 0; j < N_ / 16; ++j) {
    const int jb = j * 16;
    // K tile as 16 B-fragments (4x16 f32 layout): lane holds K[jb+lo][4t+2hi+{0,1}]
    const float* kp = Kb + (size_t)(jb + lo) * D_ + 2 * hi;
    v2f kf[16];
#pragma unroll
    for (int t = 0; t < 16; ++t)
      kf[t] = *(const v2f*)(kp + 4 * t);

    v8f s = {0.f, 0.f, 0.f, 0.f, 0.f, 0.f, 0.f, 0.f};
#pragma unroll
    for (int t = 0; t < 16; ++t)
      s = __builtin_amdgcn_wmma_f32_16x16x4_f32(false, qf[t], false, kf[t],
                                                (short)0, s, false, false);

    // C layout: this lane owns column key = jb+lo, rows m = 8*hi + r
    const int key = jb + lo;
    const int mk  = mask[(size_t)b * N_ + key];
    float sv[8];
#pragma unroll
    for (int r = 0; r < 8; ++r) {
      float x = s[r] * SCALE;
      sv[r] = mk ? NEG_BIG : x;
      Abase[(size_t)(8 * hi + r) * N_ + key] = sv[r];   // raw masked score
    }

    // per-row reductions across the 16 lanes of each half-wave
#pragma unroll
    for (int r = 0; r < 8; ++r) {
      float tm = sv[r];
      tm = fmaxf(tm, __shfl_xor(tm, 1, 32));
      tm = fmaxf(tm, __shfl_xor(tm, 2, 32));
      tm = fmaxf(tm, __shfl_xor(tm, 4, 32));
      tm = fmaxf(tm, __shfl_xor(tm, 8, 32));
      float e = __expf(sv[r] - tm);
      float ts = e;
      ts += __shfl_xor(ts, 1, 32);
      ts += __shfl_xor(ts, 2, 32);
      ts += __shfl_xor(ts, 4, 32);
      ts += __shfl_xor(ts, 8, 32);
      // online softmax update (all finite: NEG_BIG sentinel, never -inf)
      float nm = fmaxf(Mv[r], tm);
      Lv[r] = Lv[r] * __expf(Mv[r] - nm) + ts * __expf(tm - nm);
      Mv[r] = nm;
    }
  }

  // ---------- redistribute per-row (M, L) to A-fragment layout via LDS ----------
  if (lo == 0) {                 // lanes 0 and 16: rows 0-7 / 8-15
#pragma unroll
    for (int r = 0; r < 8; ++r) {
      red[w][8 * hi + r]      = Mv[r];
      red[w][16 + 8 * hi + r] = Lv[r];
    }
  }
  __syncthreads();               // also fences pass-1 global stores WG-wide
  const float Mrow = red[w][lo];
  const float invL = 1.0f / red[w][16 + lo];

  // ---------- pass 2: p = exp(s-M)/L (write attn), X += p * V via WMMA ----------
  v8f acc[4];
#pragma unroll
  for (int c = 0; c < 4; ++c)
    acc[c] = (v8f){0.f, 0.f, 0.f, 0.f, 0.f, 0.f, 0.f, 0.f};

  // this lane's attn row in A-fragment layout: row = qbase + lo
  float* arp = Aout + ((size_t)bh * N_ + qbase + lo) * N_;

  for (int j = 0; j < N_ / 16; ++j) {
    const int jb = j * 16;
#pragma unroll
    for (int t = 0; t < 4; ++t) {
      const int k0 = jb + 4 * t + 2 * hi;       // first of this lane's 2 keys
      v2f sf = *(const v2f*)(arp + k0);
      v2f p;
      p.x = __expf(sf.x - Mrow) * invL;
      p.y = __expf(sf.y - Mrow) * invL;
      *(v2f*)(arp + k0) = p;                    // final attn probabilities

      // V B-fragments: bf = { V[k0][col], V[k0+1][col] }, col = 16*ct + lo
      const float* vp = Vb + (size_t)k0 * D_ + lo;
#pragma unroll
      for (int ct = 0; ct < 4; ++ct) {
        v2f bf;
        bf.x = vp[ct * 16];
        bf.y = vp[ct * 16 + D_];
        acc[ct] = __builtin_amdgcn_wmma_f32_16x16x4_f32(false, p, false, bf,
                                                        (short)0, acc[ct],
                                                        false, false);
      }
    }
  }

  // ---------- store X (C layout: row = qbase + 8*hi + r, col = 16*ct + lo) ----------
  float* xp = Xout + ((size_t)bh * N_ + qbase + 8 * hi) * D_ + lo;
#pragma unroll
  for (int ct = 0; ct < 4; ++ct)
#pragma unroll
    for (int r = 0; r < 8; ++r)
      xp[(size_t)r * D_ + ct * 16] = acc[ct][r];
}

extern "C" void kernel_launch(void* const* d_in, const int* in_sizes, int n_in,
                              void* d_out, int out_size, void* d_ws, size_t ws_size,
                              hipStream_t stream) {
  const float* Q    = (const float*)d_in[0];
  const float* K    = (const float*)d_in[1];
  const float* V    = (const float*)d_in[2];
  const int*   mask = (const int*)d_in[3];

  float* Xout = (float*)d_out;                                // [B,H,N,D]
  float* Aout = Xout + (size_t)B_ * H_ * N_ * D_;             // [B,H,N,N]

  dim3 grid(B_ * H_ * (N_ / 128));   // 512 WGs
  dim3 block(256);                   // 8 waves, 16 q-rows each
  hipLaunchKernelGGL(attn_fp32_wmma_kernel, grid, block, 0, stream,
                     Q, K, V, mask, Xout, Aout);
}